// SymmetricPad2d_12421045420114
// MI455X (gfx1250) — compile-verified
//
#include <hip/hip_runtime.h>

// SymmetricPad2d: (16,64,256,256) f32 -> (16,64,262,262) f32, pad=3 all sides,
// edge-inclusive reflection:
//   out idx o:  o<3    -> in 2-o
//               3..258 -> in o-3
//               >=259  -> in 514-o
// Same mapping for rows and columns (H == W == 256).
//
// Memory-bound: ~0.55 GB traffic -> ~24 us at 23.3 TB/s. No matrix math, so no
// WMMA. Streaming working set (549 MB) > 192 MB L2 -> non-temporal (gfx1250
// TH=NT cache hint) on both loads and stores. 8 rows per block to amortize the
// per-block SALU prologue (workgroup-id reconstruction + kernarg load).

#define PAD   3
#define N_IN  256
#define N_OUT (N_IN + 2 * PAD)   // 262
#define BC    (16 * 64)          // 1024 (batch * channels)
#define ROWS_PER_BLOCK 8
#define ROW_BLOCKS ((N_OUT + ROWS_PER_BLOCK - 1) / ROWS_PER_BLOCK)  // 33

__device__ __forceinline__ int reflect_idx(int o) {
    int lo  = PAD - 1 - o;                 // o < 3
    int mid = o - PAD;                     // 3 <= o < 259
    int hi  = 2 * N_IN + PAD - 1 - o;      // 514 - o, o >= 259
    return (o < PAD) ? lo : ((o < N_IN + PAD) ? mid : hi);
}

__global__ __launch_bounds__(256)
void sympad2d_kernel(const float* __restrict__ x, float* __restrict__ out) {
    const int bc = blockIdx.y;                       // 0..1023 (batch*channel)
    const int r0 = blockIdx.x * ROWS_PER_BLOCK;      // first output row

    const float* __restrict__ ibase = x   + (size_t)bc * (N_IN  * N_IN);
    float*       __restrict__ obase = out + (size_t)bc * (N_OUT * N_OUT);

    #pragma unroll
    for (int i = 0; i < ROWS_PER_BLOCK; ++i) {
        const int ho = r0 + i;
        if (ho >= N_OUT) break;                      // only last x-block exits early
        const int hi = reflect_idx(ho);              // uniform -> SALU

        const float* __restrict__ irow = ibase + (size_t)hi * N_IN;
        float*       __restrict__ orow = obase + (size_t)ho * N_OUT;

        // 262 columns, 256 threads: iter 0 -> co 0..255, iter 1 -> co 256..261.
        // Consecutive lanes write consecutive output dwords (fully coalesced);
        // interior loads are consecutive too (shifted by -3).
        for (int co = threadIdx.x; co < N_OUT; co += 256) {
            const int ci = reflect_idx(co);
            const float v = __builtin_nontemporal_load(irow + ci);
            __builtin_nontemporal_store(v, orow + co);
        }
    }
}

extern "C" void kernel_launch(void* const* d_in, const int* in_sizes, int n_in,
                              void* d_out, int out_size, void* d_ws, size_t ws_size,
                              hipStream_t stream) {
    const float* x  = (const float*)d_in[0];
    float*      out = (float*)d_out;

    dim3 grid(ROW_BLOCKS, BC);   // (33, 1024)
    dim3 block(256);
    sympad2d_kernel<<<grid, block, 0, stream>>>(x, out);
}